// TemporalGNN_88003879895451
// MI455X (gfx1250) — compile-verified
//
#include <hip/hip_runtime.h>
#include <cmath>

#define NN 50000            // nodes
#define NE 800000           // edges
#define NB 4                // batch
#define NF 32               // feature width
#define NM (NB * NN)        // 200000 rows (B*N), divisible by 16

typedef __attribute__((ext_vector_type(2))) float v2f;
typedef __attribute__((ext_vector_type(8))) float v8f;

// ---- f32 WMMA 16x16x4 wrapper (D = A(16x4) * B(4x16) + C) ----
__device__ __forceinline__ v8f wmma4(v2f a, v2f b, v8f c) {
    return __builtin_amdgcn_wmma_f32_16x16x4_f32(
        /*neg_a=*/false, a, /*neg_b=*/false, b,
        /*c_mod=*/(short)0, c, /*reuse_a=*/false, /*reuse_b=*/false);
}

__global__ void k_zero(float* __restrict__ p, int n) {
    int t = blockIdx.x * blockDim.x + threadIdx.x;
    if (t < n) p[t] = 0.0f;
}

__global__ void k_deg_count(const long long* __restrict__ dst,
                            float* __restrict__ deg) {
    int e = blockIdx.x * blockDim.x + threadIdx.x;
    if (e < NE) atomicAdd(&deg[(int)dst[e]], 1.0f);
}

// deg buffer holds raw counts; finalize to inv_deg (in place) + dis = rsqrt(deg+1)
__global__ void k_deg_fin(float* __restrict__ invd, float* __restrict__ dis) {
    int n = blockIdx.x * blockDim.x + threadIdx.x;
    if (n < NN) {
        float d = invd[n] + 1.0f;    // +1 self loop
        invd[n] = 1.0f / d;
        dis[n]  = rsqrtf(d);
    }
}

__global__ void k_norm(const long long* __restrict__ src,
                       const long long* __restrict__ dst,
                       const float* __restrict__ dis,
                       float* __restrict__ norm) {
    int e = blockIdx.x * blockDim.x + threadIdx.x;
    if (e < NE) norm[e] = dis[(int)src[e]] * dis[(int)dst[e]];
}

// cell1 xw: Fin=1 -> outer product. Wc: (3,1,32); gate0 at +0, gate2 at +64.
__global__ void k_xw1(const float* __restrict__ x, const float* __restrict__ Wc,
                      float* __restrict__ xwz, float* __restrict__ xwh) {
    int t = blockIdx.x * blockDim.x + threadIdx.x;
    if (t < NM * NF) {
        int r = t >> 5, f = t & 31;
        float xv = x[r];
        xwz[t] = xv * Wc[f];
        xwh[t] = xv * Wc[64 + f];
    }
}

// Edge scatter for both gates: agg[b,dst,f] += xw[b,src,f] * norm[e]
__global__ void k_scatter(const long long* __restrict__ src,
                          const long long* __restrict__ dst,
                          const float* __restrict__ norm,
                          const float* __restrict__ xwz,
                          const float* __restrict__ xwh,
                          float* __restrict__ aggz,
                          float* __restrict__ aggh) {
    int t = blockIdx.x * blockDim.x + threadIdx.x;
    if (t >= NE * NF) return;
    int e = t >> 5, f = t & 31;
    int s = (int)src[e];
    int d = (int)dst[e];
    float w = norm[e];
#pragma unroll
    for (int b = 0; b < NB; ++b) {
        int so = (b * NN + s) * NF + f;
        int dofs = (b * NN + d) * NF + f;
        atomicAdd(&aggz[dofs], xwz[so] * w);
        atomicAdd(&aggh[dofs], xwh[so] * w);
    }
}

// cell2 xw via WMMA: xwz = h @ Wc2[0], xwh = h @ Wc2[2]; Wc2: (3,32,32)
__global__ void __launch_bounds__(256) k_xw2(const float* __restrict__ h,
                                             const float* __restrict__ Wc2,
                                             float* __restrict__ xwz,
                                             float* __restrict__ xwh) {
    int wave = threadIdx.x >> 5, lane = threadIdx.x & 31;
    int rowBase = (blockIdx.x * 8 + wave) * 16;
    if (rowBase >= NM) return;                      // wave-uniform: EXEC stays full
    int c0 = lane & 15;                             // A row sel / B-D column sel
    int khalf = (lane >> 4) << 1;                   // 0 or 2
    const float* Wz = Wc2;
    const float* Wh = Wc2 + 2 * NF * NF;
    const float* arow = h + (size_t)(rowBase + c0) * NF;
    v8f az0 = {}, az1 = {}, ah0 = {}, ah1 = {};
#pragma unroll
    for (int kk = 0; kk < 8; ++kk) {
        int k = kk * 4 + khalf;
        v2f a = *(const v2f*)(arow + k);
        v2f bz0 = { Wz[k * NF + c0],      Wz[(k + 1) * NF + c0] };
        v2f bz1 = { Wz[k * NF + c0 + 16], Wz[(k + 1) * NF + c0 + 16] };
        v2f bh0 = { Wh[k * NF + c0],      Wh[(k + 1) * NF + c0] };
        v2f bh1 = { Wh[k * NF + c0 + 16], Wh[(k + 1) * NF + c0 + 16] };
        az0 = wmma4(a, bz0, az0);
        az1 = wmma4(a, bz1, az1);
        ah0 = wmma4(a, bh0, ah0);
        ah1 = wmma4(a, bh1, ah1);
    }
    int rowOff = (lane >> 4) << 3;                  // 0 or 8
#pragma unroll
    for (int j = 0; j < 8; ++j) {
        size_t row = (size_t)(rowBase + j + rowOff);
        xwz[row * NF + c0]      = az0[j];
        xwz[row * NF + c0 + 16] = az1[j];
        xwh[row * NF + c0]      = ah0[j];
        xwh[row * NF + c0 + 16] = ah1[j];
    }
}

// Fused gate kernel: gcn = agg + xw*inv_deg + bc built as A-fragment on the fly,
// then Z = sigmoid(gcn_z @ Wlz + blz), Ht = tanh(gcn_h @ Wlh + blh),
// h_out = relu((1-Z)*Ht).  Wlz/Wlh are the top 32x32 of the (64,32) gate weights.
__global__ void __launch_bounds__(256) k_gates(
        const float* __restrict__ aggz, const float* __restrict__ aggh,
        const float* __restrict__ xwz,  const float* __restrict__ xwh,
        const float* __restrict__ invdeg,
        const float* __restrict__ bcz,  const float* __restrict__ bch,
        const float* __restrict__ Wlz,  const float* __restrict__ blz,
        const float* __restrict__ Wlh,  const float* __restrict__ blh,
        float* __restrict__ hout) {
    int wave = threadIdx.x >> 5, lane = threadIdx.x & 31;
    int rowBase = (blockIdx.x * 8 + wave) * 16;
    if (rowBase >= NM) return;                      // wave-uniform
    int c0 = lane & 15;
    int khalf = (lane >> 4) << 1;
    int myrow = rowBase + c0;
    float invd = invdeg[myrow % NN];
    const float* agz_r = aggz + (size_t)myrow * NF;
    const float* agh_r = aggh + (size_t)myrow * NF;
    const float* xz_r  = xwz  + (size_t)myrow * NF;
    const float* xh_r  = xwh  + (size_t)myrow * NF;
    v8f az0 = {}, az1 = {}, ah0 = {}, ah1 = {};
#pragma unroll
    for (int kk = 0; kk < 8; ++kk) {
        int k = kk * 4 + khalf;
        v2f a0 = *(const v2f*)(agz_r + k);
        v2f x0 = *(const v2f*)(xz_r + k);
        v2f a1 = *(const v2f*)(agh_r + k);
        v2f x1 = *(const v2f*)(xh_r + k);
        v2f gz, gh;
        gz.x = a0.x + x0.x * invd + bcz[k];
        gz.y = a0.y + x0.y * invd + bcz[k + 1];
        gh.x = a1.x + x1.x * invd + bch[k];
        gh.y = a1.y + x1.y * invd + bch[k + 1];
        v2f bz0 = { Wlz[k * NF + c0],      Wlz[(k + 1) * NF + c0] };
        v2f bz1 = { Wlz[k * NF + c0 + 16], Wlz[(k + 1) * NF + c0 + 16] };
        v2f bh0 = { Wlh[k * NF + c0],      Wlh[(k + 1) * NF + c0] };
        v2f bh1 = { Wlh[k * NF + c0 + 16], Wlh[(k + 1) * NF + c0 + 16] };
        az0 = wmma4(gz, bz0, az0);
        az1 = wmma4(gz, bz1, az1);
        ah0 = wmma4(gh, bh0, ah0);
        ah1 = wmma4(gh, bh1, ah1);
    }
    int rowOff = (lane >> 4) << 3;
    float bZ0 = blz[c0], bZ1 = blz[c0 + 16];
    float bH0 = blh[c0], bH1 = blh[c0 + 16];
#pragma unroll
    for (int j = 0; j < 8; ++j) {
        size_t row = (size_t)(rowBase + j + rowOff);
        float Z0 = 1.0f / (1.0f + __expf(-(az0[j] + bZ0)));
        float Z1 = 1.0f / (1.0f + __expf(-(az1[j] + bZ1)));
        float T0 = tanhf(ah0[j] + bH0);
        float T1 = tanhf(ah1[j] + bH1);
        hout[row * NF + c0]      = fmaxf((1.0f - Z0) * T0, 0.0f);
        hout[row * NF + c0 + 16] = fmaxf((1.0f - Z1) * T1, 0.0f);
    }
}

__global__ void k_out(const float* __restrict__ h, const float* __restrict__ Wout,
                      const float* __restrict__ bout, float* __restrict__ out) {
    int r = blockIdx.x * blockDim.x + threadIdx.x;
    if (r < NM) {
        float acc = bout[0];
        const float* hr = h + (size_t)r * NF;
#pragma unroll
        for (int f = 0; f < NF; ++f) acc += hr[f] * Wout[f];
        out[r] = acc;
    }
}

static inline int cdiv(long long a, long long b) { return (int)((a + b - 1) / b); }

extern "C" void kernel_launch(void* const* d_in, const int* in_sizes, int n_in,
                              void* d_out, int out_size, void* d_ws, size_t ws_size,
                              hipStream_t stream) {
    const float*     x      = (const float*)d_in[0];
    const long long* ei     = (const long long*)d_in[1];
    const float*     Wc1    = (const float*)d_in[2];   // (3,1,32)
    const float*     bc1    = (const float*)d_in[3];   // (3,32)
    const float*     Wl1    = (const float*)d_in[4];   // (3,64,32)
    const float*     bl1    = (const float*)d_in[5];   // (3,32)
    const float*     Wc2    = (const float*)d_in[7];   // (3,32,32)
    const float*     bc2    = (const float*)d_in[8];   // (3,32)
    const float*     Wl2    = (const float*)d_in[9];   // (3,64,32)
    const float*     bl2    = (const float*)d_in[10];  // (3,32)
    const float*     Wout   = (const float*)d_in[12];  // (32,1)
    const float*     bout   = (const float*)d_in[13];  // (1,)
    float*           out    = (float*)d_out;
    // att1/att2 are dead: softmax over a length-1 vector == 1.0

    const long long* src = ei;
    const long long* dst = ei + NE;

    // ---- workspace carve-up (floats, 64-elem aligned) ----
    float* ws = (float*)d_ws;
    size_t off = 0;
    auto take = [&](size_t nflt) { float* p = ws + off; off += (nflt + 63) & ~(size_t)63; return p; };
    float* invd = take(NN);            // deg counts, then 1/deg
    float* dis  = take(NN);            // rsqrt(deg)
    float* norm = take(NE);
    float* xwz  = take((size_t)NM * NF);
    float* xwh  = take((size_t)NM * NF);
    float* aggz = take((size_t)NM * NF);   // aggz & aggh contiguous: single zero pass
    float* aggh = take((size_t)NM * NF);
    float* h1   = take((size_t)NM * NF);
    float* h2   = take((size_t)NM * NF);

    const int B = 256;
    const int gN   = cdiv(NN, B);
    const int gE   = cdiv(NE, B);
    const int gMF  = cdiv((long long)NM * NF, B);
    const int gEF  = cdiv((long long)NE * NF, B);
    const int gWMMA = cdiv(NM, 128);        // 8 waves * 16 rows per block
    const int gM   = cdiv(NM, B);

    // ---- graph structure ----
    k_zero<<<gN, B, 0, stream>>>(invd, NN);
    k_deg_count<<<gE, B, 0, stream>>>(dst, invd);
    k_deg_fin<<<gN, B, 0, stream>>>(invd, dis);
    k_norm<<<gE, B, 0, stream>>>(src, dst, dis, norm);

    // ---- cell 1 (Fin=1) : z-gate uses index 0, h-candidate uses index 2 ----
    k_xw1<<<gMF, B, 0, stream>>>(x, Wc1, xwz, xwh);
    k_zero<<<2 * gMF, B, 0, stream>>>(aggz, 2 * NM * NF);   // aggz+aggh contiguous
    k_scatter<<<gEF, B, 0, stream>>>(src, dst, norm, xwz, xwh, aggz, aggh);
    k_gates<<<gWMMA, B, 0, stream>>>(aggz, aggh, xwz, xwh, invd,
                                     bc1 + 0 * NF, bc1 + 2 * NF,
                                     Wl1 + 0 * 64 * NF, bl1 + 0 * NF,
                                     Wl1 + 2 * 64 * NF, bl1 + 2 * NF, h1);

    // ---- cell 2 (Fin=32) ----
    k_xw2<<<gWMMA, B, 0, stream>>>(h1, Wc2, xwz, xwh);
    k_zero<<<2 * gMF, B, 0, stream>>>(aggz, 2 * NM * NF);
    k_scatter<<<gEF, B, 0, stream>>>(src, dst, norm, xwz, xwh, aggz, aggh);
    k_gates<<<gWMMA, B, 0, stream>>>(aggz, aggh, xwz, xwh, invd,
                                     bc2 + 0 * NF, bc2 + 2 * NF,
                                     Wl2 + 0 * 64 * NF, bl2 + 0 * NF,
                                     Wl2 + 2 * 64 * NF, bl2 + 2 * NF, h2);

    // ---- output projection ----
    k_out<<<gM, B, 0, stream>>>(h2, Wout, bout, out);
}